// LaplacianLoss_39814346834455
// MI455X (gfx1250) — compile-verified
//
#include <hip/hip_runtime.h>
#include <stdint.h>

// Problem constants (from the reference file).
#define BATCH 16
#define NVERT 100000
#define FCNT  200000
#define EDGES (3 * FCNT)   // 600000
#define KW    (BATCH * 3)  // 48 channels per vertex (batch-major transposed layout)

// Workspace layout (in 4-byte words). Total = 600000 + 100000 + 4.8M + 4.8M
// = 10.3M words = 41.2 MB.
#define OFF_W    0                        // per-face weights, FC*3 floats
#define OFF_D    (3 * FCNT)               // degree vector, N floats
#define OFF_OUTT (OFF_D + NVERT)          // transposed accumulator, N*48 floats
#define OFF_VT   (OFF_OUTT + NVERT * KW)  // transposed V, N*48 floats

typedef float v2f __attribute__((ext_vector_type(2)));
typedef float v8f __attribute__((ext_vector_type(8)));

__device__ __forceinline__ void atomAddF32(float* p, float v) {
  // Relaxed, device-scope fp32 atomic -> global_atomic_add_f32 (L2-resident dest)
  __hip_atomic_fetch_add(p, v, __ATOMIC_RELAXED, __HIP_MEMORY_SCOPE_AGENT);
}

// ---------------------------------------------------------------------------
// K0: zero d (N floats) and Outt (N*48 floats) — adjacent region, float4 wide.
// ---------------------------------------------------------------------------
__global__ void k_zero(float4* __restrict__ p, int n4) {
  int i = blockIdx.x * blockDim.x + threadIdx.x;
  if (i < n4) p[i] = make_float4(0.f, 0.f, 0.f, 0.f);
}

// ---------------------------------------------------------------------------
// K1: per-face cotangent weights + degree accumulation.
//   d[f0] += w1+w2 ; d[f1] += w0+w2 ; d[f2] += w0+w1   (3 atomics, not 6)
// ---------------------------------------------------------------------------
__global__ void k_weights(const float* __restrict__ vert,
                          const int*   __restrict__ faces,
                          float* __restrict__ W,
                          float* __restrict__ d) {
  int f = blockIdx.x * blockDim.x + threadIdx.x;
  if (f >= FCNT) return;
  int i0 = faces[3 * f + 0], i1 = faces[3 * f + 1], i2 = faces[3 * f + 2];

  float ax = vert[3 * i0], ay = vert[3 * i0 + 1], az = vert[3 * i0 + 2];
  float bx = vert[3 * i1], by = vert[3 * i1 + 1], bz = vert[3 * i1 + 2];
  float cx = vert[3 * i2], cy = vert[3 * i2 + 1], cz = vert[3 * i2 + 2];

  float d1x = bx - cx, d1y = by - cy, d1z = bz - cz;  // v2-v3
  float d2x = cx - ax, d2y = cy - ay, d2z = cz - az;  // v3-v1
  float d3x = ax - bx, d3y = ay - by, d3z = az - bz;  // v1-v2
  float l1s = d1x * d1x + d1y * d1y + d1z * d1z;
  float l2s = d2x * d2x + d2y * d2y + d2z * d2z;
  float l3s = d3x * d3x + d3y * d3y + d3z * d3z;
  float l1 = sqrtf(l1s), l2 = sqrtf(l2s), l3 = sqrtf(l3s);

  float sp = 0.5f * (l1 + l2 + l3);
  float A  = 2.0f * sqrtf(sp * (sp - l1) * (sp - l2) * (sp - l3));
  float inv = 1.0f / (4.0f * A);

  float w0 = (l2s + l3s - l1s) * inv;  // cot23
  float w1 = (l1s + l3s - l2s) * inv;  // cot31
  float w2 = (l1s + l2s - l3s) * inv;  // cot12

  W[3 * f + 0] = w0;
  W[3 * f + 1] = w1;
  W[3 * f + 2] = w2;

  atomAddF32(&d[i0], w1 + w2);
  atomAddF32(&d[i1], w0 + w2);
  atomAddF32(&d[i2], w0 + w1);
}

// ---------------------------------------------------------------------------
// K2: transpose V (B,N,3) -> Vt (N,48): each vertex's channels become one
// contiguous 192B block (edge gathers hit 2 dense blocks, L2-resident).
// ---------------------------------------------------------------------------
__global__ void k_transpose(const float* __restrict__ V, float* __restrict__ Vt) {
  int idx = blockIdx.x * blockDim.x + threadIdx.x;
  if (idx >= NVERT * KW) return;
  int n = idx / KW;
  int k = idx - n * KW;
  int b = k / 3;
  int comp = k - 3 * b;
  Vt[idx] = V[b * (NVERT * 3) + n * 3 + comp];
}

// ---------------------------------------------------------------------------
// K3: edge scatter. 8 edges per 384-thread block (48 lanes per edge).
// Edge metadata (row, col, weight) staged into LDS via the CDNA5 async path
// (GLOBAL_LOAD_ASYNC_TO_LDS_B32, ASYNCcnt), then 2 fp32 atomics per lane.
// ---------------------------------------------------------------------------
__global__ __launch_bounds__(384) void k_scatter(const int*   __restrict__ faces,
                                                 const float* __restrict__ W,
                                                 const float* __restrict__ Vt,
                                                 float*       __restrict__ Outt) {
  __shared__ int   s_r[8];
  __shared__ int   s_c[8];
  __shared__ float s_w[8];

  const int tid = threadIdx.x;
  const int eb  = blockIdx.x * 8;  // first edge of this block

  if (tid < 24) {  // wave 0, lanes 0..23: async-stage 24 dwords of metadata
    int lane = tid & 7;
    int grp  = tid >> 3;  // 0: rows, 1: cols, 2: weights
    int e = eb + lane;
    int f = e / 3;
    int j = e - 3 * f;
    const void* gptr;
    unsigned    lds;
    if (grp == 0) {
      gptr = &faces[3 * f + ((j + 1) % 3)];
      lds  = (unsigned)(uintptr_t)&s_r[lane];
    } else if (grp == 1) {
      gptr = &faces[3 * f + ((j + 2) % 3)];
      lds  = (unsigned)(uintptr_t)&s_c[lane];
    } else {
      gptr = &W[e];
      lds  = (unsigned)(uintptr_t)&s_w[lane];
    }
    unsigned long long ga = (unsigned long long)(uintptr_t)gptr;
    asm volatile("global_load_async_to_lds_b32 %0, %1, off"
                 :: "v"(lds), "v"(ga)
                 : "memory");
    asm volatile("s_wait_asynccnt 0" ::: "memory");
  }
  __syncthreads();

  int el = tid / KW;        // 0..7: edge within block
  int k  = tid - el * KW;   // 0..47: channel
  int   r = s_r[el];
  int   c = s_c[el];
  float w = s_w[el];

  float* outr = Outt + r * KW + k;
  float* outc = Outt + c * KW + k;
  __builtin_prefetch(outr, 0, 3);  // global_prefetch_b8: warm atomic dest line

  float vr = Vt[r * KW + k];
  float vc = Vt[c * KW + k];
  atomAddF32(outr, w * vc);
  atomAddF32(outc, w * vr);
}

// ---------------------------------------------------------------------------
// K4 (WMMA): diagonal term on the matrix pipe, in the transposed domain.
//   Outt(16x48 tile) += diag(-d_16) * Vt(16x48 tile)
// diag(-d) is sliced into four 16x4 chunks => 4 chained V_WMMA_F32_16X16X4_F32
// per 16-channel group (12 WMMAs per tile, 75000 total). One wave per tile,
// grid is exact (6250 tiles) so EXEC is all-ones as WMMA requires.
// Operand layouts per ISA 7.12.2:
//   A 16x4:  VGPR0 = K{0|2} by lane-half, VGPR1 = K{1|3}
//   B 4x16:  VGPR0 = rows K{0|2}, VGPR1 = rows K{1|3} (row striped over lanes)
//   C/D 16x16: VGPRj = row j (lanes 0-15) / row j+8 (lanes 16-31)
// ---------------------------------------------------------------------------
__global__ __launch_bounds__(32) void k_diag_wmma(const float* __restrict__ Vt,
                                                  const float* __restrict__ d,
                                                  float* __restrict__ Outt) {
  const int n0   = blockIdx.x * 16;  // vertex tile base (6250 blocks, 1 wave each)
  const int lane = threadIdx.x;      // 0..31
  const int m    = lane & 15;
  const int hi   = lane >> 4;

  const float negd = -d[n0 + m];

  v2f A[4];
#pragma unroll
  for (int t = 0; t < 4; ++t) {
    A[t][0] = (m == 4 * t + 2 * hi)     ? negd : 0.0f;  // A[m, K=2*hi]
    A[t][1] = (m == 4 * t + 2 * hi + 1) ? negd : 0.0f;  // A[m, K=2*hi+1]
  }

#pragma unroll
  for (int g = 0; g < 3; ++g) {
    const int ch0 = 16 * g;

    v8f acc;  // C tile: the edge-scatter accumulator
#pragma unroll
    for (int j = 0; j < 8; ++j)
      acc[j] = Outt[(n0 + j + 8 * hi) * KW + ch0 + m];

#pragma unroll
    for (int t = 0; t < 4; ++t) {
      const int row = n0 + 4 * t + 2 * hi;
      v2f Bt;
      Bt[0] = Vt[row * KW + ch0 + m];        // B row K=2*hi of chunk t
      Bt[1] = Vt[(row + 1) * KW + ch0 + m];  // B row K=2*hi+1
      acc = __builtin_amdgcn_wmma_f32_16x16x4_f32(false, A[t], false, Bt,
                                                  (short)0, acc, false, false);
    }

#pragma unroll
    for (int j = 0; j < 8; ++j)
      Outt[(n0 + j + 8 * hi) * KW + ch0 + m] = acc[j];
  }
}

// ---------------------------------------------------------------------------
// K5: layout transform out[b,n,c] = Outt[n, 3b+c]. Coalesced HBM writes,
// strided reads serviced from L2 (Outt is 19.2MB, fully L2-resident).
// ---------------------------------------------------------------------------
__global__ void k_untranspose(const float* __restrict__ T, float* __restrict__ out) {
  int idx = blockIdx.x * blockDim.x + threadIdx.x;
  if (idx >= BATCH * NVERT * 3) return;
  int b   = idx / (NVERT * 3);
  int rem = idx - b * (NVERT * 3);
  int n    = rem / 3;
  int comp = rem - 3 * n;
  out[idx] = T[n * KW + 3 * b + comp];
}

extern "C" void kernel_launch(void* const* d_in, const int* in_sizes, int n_in,
                              void* d_out, int out_size, void* d_ws, size_t ws_size,
                              hipStream_t stream) {
  const float* V     = (const float*)d_in[0];  // (16, 100000, 3) f32
  const float* vert  = (const float*)d_in[1];  // (100000, 3) f32
  const int*   faces = (const int*)d_in[2];    // (200000, 3) i32
  float*       out   = (float*)d_out;          // (16, 100000, 3) f32

  float* ws   = (float*)d_ws;
  float* W    = ws + OFF_W;
  float* dv   = ws + OFF_D;
  float* Outt = ws + OFF_OUTT;
  float* Vt   = ws + OFF_VT;

  (void)in_sizes; (void)n_in; (void)out_size; (void)ws_size;

  // K0: zero [d | Outt] = N*49 floats (divisible by 4), float4-wide.
  int n4 = (NVERT * (KW + 1)) / 4;
  k_zero<<<(n4 + 255) / 256, 256, 0, stream>>>((float4*)dv, n4);

  // K1: per-face weights + degree vector.
  k_weights<<<(FCNT + 255) / 256, 256, 0, stream>>>(vert, faces, W, dv);

  // K2: V -> Vt transpose (batch-major 48-channel rows).
  k_transpose<<<(NVERT * KW + 255) / 256, 256, 0, stream>>>(V, Vt);

  // K3: edge scatter with async-LDS metadata staging. EDGES/8 = 75000 blocks.
  k_scatter<<<EDGES / 8, 384, 0, stream>>>(faces, W, Vt, Outt);

  // K4: WMMA diagonal update, in place on Outt. NVERT/16 = 6250 tiles.
  k_diag_wmma<<<NVERT / 16, 32, 0, stream>>>(Vt, dv, Outt);

  // K5: final layout transform.
  k_untranspose<<<(BATCH * NVERT * 3 + 255) / 256, 256, 0, stream>>>(Outt, out);
}